// CausalSelfAttention_30700426231921
// MI455X (gfx1250) — compile-verified
//
#include <hip/hip_runtime.h>
#include <math.h>
#include <stdint.h>

// ---------------------------------------------------------------------------
// CausalSelfAttention with interval bounds, fp32 WMMA + async-LDS (gfx1250).
// B=2, T=1024, C=768, nh=12, hs=64.
// ---------------------------------------------------------------------------

typedef __attribute__((ext_vector_type(2))) float v2f;
typedef __attribute__((ext_vector_type(8))) float v8f;
typedef int v4i __attribute__((vector_size(4 * sizeof(int))));

#define TB  256
#define BM  128          // macro tile rows (8 waves x 16)
#define BN  64           // macro tile cols
#define BKK 32           // K step
#define BKP 36           // padded LDS row (144B: 16B-aligned, conflict-free)
#define HSD 64

#define AS1 __attribute__((address_space(1)))
#define AS3 __attribute__((address_space(3)))

// ---- gfx1250 async global->LDS copies (ASYNCcnt) with safe fallbacks -------
#if defined(__AMDGCN__) && __has_builtin(__builtin_amdgcn_global_load_async_to_lds_b32) && __has_builtin(__builtin_amdgcn_global_load_async_to_lds_b128)
#define HAVE_ASYNC_LDS 1
#endif

__device__ __forceinline__ void g2l_b32(const float* g, float* l)
{
#if defined(HAVE_ASYNC_LDS)
    __builtin_amdgcn_global_load_async_to_lds_b32(
        (AS1 int*)(uintptr_t)g, (AS3 int*)(uint32_t)(uintptr_t)l, 0, 0);
#else
    *l = *g;
#endif
}

__device__ __forceinline__ void g2l_b128(const float* g, float* l)
{
#if defined(HAVE_ASYNC_LDS)
    __builtin_amdgcn_global_load_async_to_lds_b128(
        (AS1 v4i*)(uintptr_t)g, (AS3 v4i*)(uint32_t)(uintptr_t)l, 0, 0);
#else
    l[0] = g[0]; l[1] = g[1]; l[2] = g[2]; l[3] = g[3];
#endif
}

__device__ __forceinline__ void wait_async_lds()
{
#if defined(__AMDGCN__)
#if __has_builtin(__builtin_amdgcn_s_wait_asynccnt)
    __builtin_amdgcn_s_wait_asynccnt(0);
#else
    asm volatile("s_wait_asynccnt 0x0" ::: "memory");
#endif
#endif
}

// ---------------------------------------------------------------------------
// Generic batched GEMM, D = alpha * Acomb @ opB + beta*D
//   Acomb: amode 0 -> A ; 1 -> 0.5*(A+A2) ; 2 -> 0.5*(A2-A)
//   opB:   transB=1 -> B is [N x K] row-major; transB=0 -> B is [K x N]
//          babs=1  -> |B| elementwise
//   batch z: operand offsets = (z/nh)*s?b + (z%nh)*s?h
//   causal=1: skip tiles entirely above the diagonal
// 256 threads = 8 waves; macro tile 128x64; wave computes 16x64 with four
// v_wmma_f32_16x16x4_f32 accumulators. Double-buffered LDS; pure-copy
// operands staged with global_load_async_to_lds (overlap with WMMA).
// ---------------------------------------------------------------------------
__global__ __launch_bounds__(TB)
void gemm_wmma_f32(const float* __restrict__ A, const float* __restrict__ A2, int amode,
                   const float* __restrict__ Bm, int transB, int babs,
                   float* __restrict__ D,
                   int M, int N, int K,
                   int lda, int ldb, int ldd,
                   float alpha, int betaflag,
                   int nh,
                   long long sAb, long long sAh,
                   long long sBb, long long sBh,
                   long long sDb, long long sDh,
                   int causal)
{
    __shared__ float As[2][BM][BKP];
    __shared__ float Bs[2][BN][BKP];

    const int m0 = blockIdx.y * BM;
    const int n0 = blockIdx.x * BN;
    if (causal && n0 > m0 + (BM - 1)) return;

    const int z = blockIdx.z;
    const long long offA = (long long)(z / nh) * sAb + (long long)(z % nh) * sAh;
    const long long offB = (long long)(z / nh) * sBb + (long long)(z % nh) * sBh;
    const long long offD = (long long)(z / nh) * sDb + (long long)(z % nh) * sDh;
    A += offA;
    if (A2) A2 += offA;
    Bm += offB;
    D += offD;

    const int tid   = threadIdx.x;
    const int wave  = tid >> 5;
    const int lane  = tid & 31;
    const int l16   = lane & 15;
    const int khalf = (lane >> 4) << 1;   // 0 or 2

    const int wm = wave * 16;             // wave row offset in tile

    v8f c0 = {}, c1 = {}, c2 = {}, c3 = {};

    const int nsteps = K / BKK;

    // ---- tile staging ------------------------------------------------------
    auto stage = [&](int buf, int k0) {
        // A tile: As[buf][r][kk]
        if (amode == 0) {
            #pragma unroll
            for (int i = 0; i < (BM * BKK) / (TB * 4); ++i) {   // 4 x b128
                int idx = tid + i * TB;
                int r  = idx >> 3;
                int c4 = (idx & 7) << 2;
                g2l_b128(A + (long long)(m0 + r) * lda + (k0 + c4),
                         &As[buf][r][c4]);
            }
        } else {
            #pragma unroll
            for (int i = 0; i < (BM * BKK) / TB; ++i) {
                int idx = tid + i * TB;
                int r  = idx >> 5;
                int cc = idx & 31;
                long long g = (long long)(m0 + r) * lda + (k0 + cc);
                float av = (amode == 1) ? 0.5f * (A[g] + A2[g])
                                        : 0.5f * (A2[g] - A[g]);
                As[buf][r][cc] = av;
            }
        }
        // B tile: Bs[buf][n][kk]
        if (transB) {
            if (!babs) {
                #pragma unroll
                for (int i = 0; i < (BN * BKK) / (TB * 4); ++i) {  // 2 x b128
                    int idx = tid + i * TB;
                    int r  = idx >> 3;
                    int c4 = (idx & 7) << 2;
                    g2l_b128(Bm + (long long)(n0 + r) * ldb + (k0 + c4),
                             &Bs[buf][r][c4]);
                }
            } else {
                #pragma unroll
                for (int i = 0; i < (BN * BKK) / TB; ++i) {
                    int idx = tid + i * TB;
                    int r  = idx >> 5;
                    int cc = idx & 31;
                    Bs[buf][r][cc] =
                        fabsf(Bm[(long long)(n0 + r) * ldb + (k0 + cc)]);
                }
            }
        } else {  // B is [K x N]: LDS column writes -> per-dword
            #pragma unroll
            for (int i = 0; i < (BN * BKK) / TB; ++i) {
                int idx = tid + i * TB;
                int cc = idx >> 6;        // kk
                int r  = idx & 63;        // n (consecutive tid -> coalesced)
                const float* src = Bm + (long long)(k0 + cc) * ldb + (n0 + r);
                if (!babs) g2l_b32(src, &Bs[buf][r][cc]);
                else       Bs[buf][r][cc] = fabsf(*src);
            }
        }
    };

    stage(0, 0);

    for (int s = 0; s < nsteps; ++s) {
        const int cur = s & 1;
        wait_async_lds();
        __syncthreads();
        if (s + 1 < nsteps) stage(1 - cur, (s + 1) * BKK);

        #pragma unroll
        for (int kk = 0; kk < BKK; kk += 4) {
            const int kb = kk + khalf;
            // A frag 16x4: lane l -> (M = l%16, K = khalf + {0,1})
            v2f a;
            a.x = As[cur][wm + l16][kb];
            a.y = As[cur][wm + l16][kb + 1];
            v2f b0; b0.x = Bs[cur][l16][kb];      b0.y = Bs[cur][l16][kb + 1];
            v2f b1; b1.x = Bs[cur][16 + l16][kb]; b1.y = Bs[cur][16 + l16][kb + 1];
            v2f b2; b2.x = Bs[cur][32 + l16][kb]; b2.y = Bs[cur][32 + l16][kb + 1];
            v2f b3; b3.x = Bs[cur][48 + l16][kb]; b3.y = Bs[cur][48 + l16][kb + 1];
            c0 = __builtin_amdgcn_wmma_f32_16x16x4_f32(false, a, false, b0, (short)0, c0, false, false);
            c1 = __builtin_amdgcn_wmma_f32_16x16x4_f32(false, a, false, b1, (short)0, c1, false, false);
            c2 = __builtin_amdgcn_wmma_f32_16x16x4_f32(false, a, false, b2, (short)0, c2, false, false);
            c3 = __builtin_amdgcn_wmma_f32_16x16x4_f32(false, a, false, b3, (short)0, c3, false, false);
        }
    }

    // ---- epilogue. C/D: VGPR r, lanes0-15 -> M=r, lanes16-31 -> M=r+8, N=lane%16
    const int rowhalf = (lane >> 4) * 8;
    #pragma unroll
    for (int r = 0; r < 8; ++r) {
        const int row = m0 + wm + r + rowhalf;
        long long d0 = (long long)row * ldd + (n0 + l16);
        float v0 = alpha * c0[r];
        float v1 = alpha * c1[r];
        float v2 = alpha * c2[r];
        float v3 = alpha * c3[r];
        if (betaflag) {
            v0 += D[d0]; v1 += D[d0 + 16]; v2 += D[d0 + 32]; v3 += D[d0 + 48];
        }
        D[d0]      = v0;
        D[d0 + 16] = v1;
        D[d0 + 32] = v2;
        D[d0 + 48] = v3;
    }
}

// ---------------------------------------------------------------------------
// lo/hi combine for QKV:  LO = S - E + b ; HI = S + E + b ; S += b
// ---------------------------------------------------------------------------
__global__ __launch_bounds__(256)
void lohi_combine(float* __restrict__ S, const float* __restrict__ E,
                  float* __restrict__ LO, float* __restrict__ HI,
                  const float* __restrict__ bias, int n, int ldc)
{
    int idx = blockIdx.x * 256 + threadIdx.x;
    if (idx >= n) return;
    float s  = S[idx];
    float e  = E[idx];
    float bb = bias[idx % ldc];
    LO[idx] = s - e + bb;
    HI[idx] = s + e + bb;
    S[idx]  = s + bb;
}

// ---------------------------------------------------------------------------
// Exact interval bounds of (q . k) * scale over hs=64 (per-element corner
// min/max, clamped, summed) -- inherently elementwise VALU tile kernel.
// ---------------------------------------------------------------------------
__global__ __launch_bounds__(256)
void att_bounds(const float* __restrict__ S, const float* __restrict__ E,
                float* __restrict__ att_lo, float* __restrict__ att_hi,
                int T, int C3, int nh, float scale, float clampv)
{
    __shared__ float qlo[32][HSD + 1];
    __shared__ float qhi[32][HSD + 1];
    __shared__ float klo[32][HSD + 1];
    __shared__ float khi[32][HSD + 1];

    const int z  = blockIdx.z;
    const int b  = z / nh;
    const int h  = z % nh;
    const int ti = blockIdx.y * 32;
    const int tj = blockIdx.x * 32;
    if (tj > ti + 31) return;   // entirely above the diagonal -> masked

    const int tid = threadIdx.x;
    const long long qbase = ((long long)b * T + ti) * C3 + h * HSD;        // q
    const long long kbase = ((long long)b * T + tj) * C3 + 768 + h * HSD;  // k

    #pragma unroll
    for (int i = 0; i < 8; ++i) {
        int idx = tid + i * 256;
        int r = idx >> 6;
        int d = idx & 63;
        long long qi = qbase + (long long)r * C3 + d;
        float s = S[qi], e = E[qi];
        qlo[r][d] = s - e;
        qhi[r][d] = s + e;
        long long ki = kbase + (long long)r * C3 + d;
        s = S[ki]; e = E[ki];
        klo[r][d] = s - e;
        khi[r][d] = s + e;
    }
    __syncthreads();

    const int jj = tid & 31;
    const int i0 = tid >> 5;
    float accLo[4] = {0.f, 0.f, 0.f, 0.f};
    float accHi[4] = {0.f, 0.f, 0.f, 0.f};

    for (int d = 0; d < HSD; ++d) {
        const float kl = klo[jj][d];
        const float kh = khi[jj][d];
        #pragma unroll
        for (int u = 0; u < 4; ++u) {
            const int ii = i0 + u * 8;
            const float ql = qlo[ii][d];
            const float qh = qhi[ii][d];
            const float pll = ql * kl, plu = ql * kh;
            const float pul = qh * kl, puu = qh * kh;
            float pmin = fminf(fminf(pll, plu), fminf(pul, puu));
            float pmax = fmaxf(fmaxf(pll, plu), fmaxf(pul, puu));
            pmin = fminf(fmaxf(pmin, -clampv), clampv);
            pmax = fminf(fmaxf(pmax, -clampv), clampv);
            accLo[u] += pmin;
            accHi[u] += pmax;
        }
    }

    const long long obase = ((long long)z * T + ti) * T + tj;
    #pragma unroll
    for (int u = 0; u < 4; ++u) {
        const int ii = i0 + u * 8;
        long long o = obase + (long long)ii * T + jj;
        att_lo[o] = accLo[u] * scale;
        att_hi[o] = accHi[u] * scale;
    }
}

// ---------------------------------------------------------------------------
// Masked softmax + bound softmax, one row (b,h,i) per workgroup.
// ---------------------------------------------------------------------------
__device__ __forceinline__ float blk_reduce(float v, float* red, bool ismax)
{
    const int tid = threadIdx.x;
    red[tid] = v;
    __syncthreads();
    for (int s = 128; s > 0; s >>= 1) {
        if (tid < s) red[tid] = ismax ? fmaxf(red[tid], red[tid + s])
                                      : (red[tid] + red[tid + s]);
        __syncthreads();
    }
    float r = red[0];
    __syncthreads();
    return r;
}

__global__ __launch_bounds__(256)
void softmax_bounds(float* __restrict__ att, float* __restrict__ att_lo,
                    float* __restrict__ att_hi, int T)
{
    __shared__ float red[256];
    const int row = blockIdx.x;           // z*T + i
    const int i   = row & (T - 1);
    const int L   = i + 1;
    float* pa = att    + (long long)row * T;
    float* pl = att_lo + (long long)row * T;
    float* ph = att_hi + (long long)row * T;
    const int tid = threadIdx.x;

    float va[4], vl[4], vh[4];

    float m = -INFINITY;
    #pragma unroll
    for (int u = 0; u < 4; ++u) {
        int j = tid + u * 256;
        va[u] = (j < L) ? pa[j] : -INFINITY;
        m = fmaxf(m, va[u]);
    }
    m = blk_reduce(m, red, true);
    float sum = 0.f;
    #pragma unroll
    for (int u = 0; u < 4; ++u) {
        int j = tid + u * 256;
        va[u] = (j < L) ? expf(va[u] - m) : 0.f;
        sum += va[u];
    }
    sum = blk_reduce(sum, red, false);

    float dl = 0.f, dh = 0.f;
    #pragma unroll
    for (int u = 0; u < 4; ++u) {
        int j = tid + u * 256;
        vl[u] = (j < L) ? expf(pl[j]) : 0.f;
        vh[u] = (j < L) ? expf(ph[j]) : 0.f;
        dl += vl[u];
        dh += vh[u];
    }
    dl = blk_reduce(dl, red, false);
    dh = blk_reduce(dh, red, false);

    const float inv  = 1.f / sum;
    const float invl = 1.f / dl;
    const float invh = 1.f / dh;
    #pragma unroll
    for (int u = 0; u < 4; ++u) {
        int j = tid + u * 256;
        if (j < T) {
            pa[j] = va[u] * inv;
            pl[j] = vl[u] * invh;   // sm_lo = exp_lo / den_hi
            ph[j] = vh[u] * invl;   // sm_hi = exp_hi / den_lo
        }
    }
}

// ---------------------------------------------------------------------------
// Final: out += b ; out_lo = OC - OR + b ; out_hi = OC + OR + b
// ---------------------------------------------------------------------------
__global__ __launch_bounds__(256)
void final_combine(float* __restrict__ o0, float* __restrict__ o1,
                   float* __restrict__ o2,
                   const float* __restrict__ OC, const float* __restrict__ ORr,
                   const float* __restrict__ bias, int n, int ldc)
{
    int idx = blockIdx.x * 256 + threadIdx.x;
    if (idx >= n) return;
    float bb = bias[idx % ldc];
    o0[idx] += bb;
    float c = OC[idx];
    float r = ORr[idx];
    o1[idx] = c - r + bb;
    o2[idx] = c + r + bb;
}

// ---------------------------------------------------------------------------
// Host
// ---------------------------------------------------------------------------
extern "C" void kernel_launch(void* const* d_in, const int* in_sizes, int n_in,
                              void* d_out, int out_size, void* d_ws, size_t ws_size,
                              hipStream_t stream)
{
    (void)in_sizes; (void)n_in; (void)out_size; (void)ws_size;

    const int Bb = 2, T = 1024, C = 768, NH = 12, C3 = 2304;
    const int M  = Bb * T;                 // 2048
    const float scale  = 0.125f;           // 1/sqrt(64)
    const float clampv = 1000000000.0f;

    const float* x      = (const float*)d_in[0];
    const float* xerr   = (const float*)d_in[1];
    const float* W_attn = (const float*)d_in[2];
    const float* b_attn = (const float*)d_in[3];
    const float* W_proj = (const float*)d_in[4];
    const float* b_proj = (const float*)d_in[5];

    float* o0 = (float*)d_out;             // out
    float* o1 = o0 + (size_t)M * C;        // out_lo
    float* o2 = o1 + (size_t)M * C;        // out_hi

    // workspace layout (floats)
    float* S   = (float*)d_ws;                                   // [M, 2304]
    float* E   = S   + (size_t)M * C3;                           // [M, 2304]
    float* LOq = E   + (size_t)M * C3;                           // [M, 2304]
    float* HIq = LOq + (size_t)M * C3;                           // [M, 2304]
    float* ATT = HIq + (size_t)M * C3;                           // [B,NH,T,T]
    float* ATL = ATT + (size_t)Bb * NH * T * T;
    float* ATH = ATL + (size_t)Bb * NH * T * T;
    float* Y   = ATH + (size_t)Bb * NH * T * T;                  // [M, C]
    float* YL  = Y   + (size_t)M * C;
    float* YH  = YL  + (size_t)M * C;
    float* OC  = YH  + (size_t)M * C;
    float* ORr = OC  + (size_t)M * C;

    const dim3 blk(TB);
    const long long zero = 0;

    // 1) S = x @ W_attn^T
    gemm_wmma_f32<<<dim3(C3 / BN, M / BM, 1), blk, 0, stream>>>(
        x, nullptr, 0, W_attn, 1, 0, S, M, C3, C, C, C, C3,
        1.0f, 0, 1, zero, zero, zero, zero, zero, zero, 0);
    // 2) E = x_error @ |W_attn|^T
    gemm_wmma_f32<<<dim3(C3 / BN, M / BM, 1), blk, 0, stream>>>(
        xerr, nullptr, 0, W_attn, 1, 1, E, M, C3, C, C, C, C3,
        1.0f, 0, 1, zero, zero, zero, zero, zero, zero, 0);
    // 3) LO = S-E+b, HI = S+E+b, S += b
    {
        int n = M * C3;
        lohi_combine<<<dim3((n + 255) / 256), blk, 0, stream>>>(S, E, LOq, HIq, b_attn, n, C3);
    }
    // 4) att = q @ k^T * scale (batched, causal tile skip)
    gemm_wmma_f32<<<dim3(T / BN, T / BM, Bb * NH), blk, 0, stream>>>(
        S, nullptr, 0, S + 768, 1, 0, ATT, T, T, HSD, C3, C3, T,
        scale, 0, NH,
        (long long)T * C3, (long long)HSD,
        (long long)T * C3, (long long)HSD,
        (long long)NH * T * T, (long long)T * T, 1);
    // 5) exact interval bounds of q.k (VALU tile kernel)
    att_bounds<<<dim3(T / 32, T / 32, Bb * NH), blk, 0, stream>>>(
        S, E, ATL, ATH, T, C3, NH, scale, clampv);
    // 6) softmax + bound softmax (in place)
    softmax_bounds<<<dim3(Bb * NH * T), blk, 0, stream>>>(ATT, ATL, ATH, T);

    // 7) y = sm @ v
    gemm_wmma_f32<<<dim3(HSD / BN, T / BM, Bb * NH), blk, 0, stream>>>(
        ATT, nullptr, 0, S + 1536, 0, 0, Y, T, HSD, T, T, C3, C,
        1.0f, 0, NH,
        (long long)NH * T * T, (long long)T * T,
        (long long)T * C3, (long long)HSD,
        (long long)T * C, (long long)HSD, 0);
    // 8) y_lo = smc @ v_lo - smr @ |v_lo|
    gemm_wmma_f32<<<dim3(HSD / BN, T / BM, Bb * NH), blk, 0, stream>>>(
        ATL, ATH, 1, LOq + 1536, 0, 0, YL, T, HSD, T, T, C3, C,
        1.0f, 0, NH,
        (long long)NH * T * T, (long long)T * T,
        (long long)T * C3, (long long)HSD,
        (long long)T * C, (long long)HSD, 0);
    gemm_wmma_f32<<<dim3(HSD / BN, T / BM, Bb * NH), blk, 0, stream>>>(
        ATL, ATH, 2, LOq + 1536, 0, 1, YL, T, HSD, T, T, C3, C,
        -1.0f, 1, NH,
        (long long)NH * T * T, (long long)T * T,
        (long long)T * C3, (long long)HSD,
        (long long)T * C, (long long)HSD, 0);
    // 9) y_hi = smc @ v_hi + smr @ |v_hi|
    gemm_wmma_f32<<<dim3(HSD / BN, T / BM, Bb * NH), blk, 0, stream>>>(
        ATL, ATH, 1, HIq + 1536, 0, 0, YH, T, HSD, T, T, C3, C,
        1.0f, 0, NH,
        (long long)NH * T * T, (long long)T * T,
        (long long)T * C3, (long long)HSD,
        (long long)T * C, (long long)HSD, 0);
    gemm_wmma_f32<<<dim3(HSD / BN, T / BM, Bb * NH), blk, 0, stream>>>(
        ATL, ATH, 2, HIq + 1536, 0, 1, YH, T, HSD, T, T, C3, C,
        1.0f, 1, NH,
        (long long)NH * T * T, (long long)T * T,
        (long long)T * C3, (long long)HSD,
        (long long)T * C, (long long)HSD, 0);

    // 10) out = y @ W_proj^T (bias added later)
    gemm_wmma_f32<<<dim3(C / BN, M / BM, 1), blk, 0, stream>>>(
        Y, nullptr, 0, W_proj, 1, 0, o0, M, C, C, C, C, C,
        1.0f, 0, 1, zero, zero, zero, zero, zero, zero, 0);
    // 11) OC = yc @ W_proj^T ; OR = yr @ |W_proj|^T
    gemm_wmma_f32<<<dim3(C / BN, M / BM, 1), blk, 0, stream>>>(
        YL, YH, 1, W_proj, 1, 0, OC, M, C, C, C, C, C,
        1.0f, 0, 1, zero, zero, zero, zero, zero, zero, 0);
    gemm_wmma_f32<<<dim3(C / BN, M / BM, 1), blk, 0, stream>>>(
        YL, YH, 2, W_proj, 1, 1, ORr, M, C, C, C, C, C,
        1.0f, 0, 1, zero, zero, zero, zero, zero, zero, 0);
    // 12) biases + lo/hi assembly
    {
        int n = M * C;
        final_combine<<<dim3((n + 255) / 256), blk, 0, stream>>>(o0, o1, o2, OC, ORr, b_proj, n, C);
    }
}